// GAT_3315714753148
// MI455X (gfx1250) — compile-verified
//
#include <hip/hip_runtime.h>
#include <hip/hip_bf16.h>

typedef __attribute__((ext_vector_type(2))) float    v2f;
typedef __attribute__((ext_vector_type(8))) float    v8f;
typedef __attribute__((ext_vector_type(4))) unsigned v4u;
typedef __attribute__((ext_vector_type(8))) int      v8i;
typedef __attribute__((ext_vector_type(4))) int      v4i;

#define N_NODES  20000
#define N_EDGES  320000
#define N_GRAPHS 64

// ---------------- ordered-float helpers (atomic max on float incl. negatives) --------
__device__ __forceinline__ unsigned f2ord(float f) {
    unsigned u = __float_as_uint(f);
    return (u & 0x80000000u) ? ~u : (u | 0x80000000u);
}
__device__ __forceinline__ float ord2f(unsigned o) {
    return (o & 0x80000000u) ? __uint_as_float(o & 0x7FFFFFFFu) : __uint_as_float(~o);
}
#define ORD_NEG_INF 0x007FFFFFu  // f2ord(-inf)

// ---------------- fill kernels ----------------
__global__ void fill_u32_kernel(unsigned* __restrict__ p, long n, unsigned v) {
    long i = (long)blockIdx.x * blockDim.x + threadIdx.x;
    if (i < n) p[i] = v;
}
__global__ void fill_f32_kernel(float* __restrict__ p, long n, float v) {
    long i = (long)blockIdx.x * blockDim.x + threadIdx.x;
    if (i < n) p[i] = v;
}

// ---------------- TDM 2-D tile load: global -> LDS (D# per ISA 08 §8.3-8.6) --------
// data_size=4B, count=1, type=2("image"), 2-D tile (groups 2/3 zero, tile_dim2=0).
__device__ __forceinline__ void tdm_load_2d(
    unsigned lds_addr, const void* gptr,
    unsigned tensor_d0, unsigned tensor_d1,      // extents from tile start (elements)
    unsigned tile_d0, unsigned tile_d1,          // tile dims (elements)
    unsigned long long stride0,                  // row stride (elements)
    unsigned pad_interval_code, unsigned pad_amount_code)
{
    const unsigned long long ga = (unsigned long long)gptr;
    v4u g0;
    g0.x = 1u;                                        // count=1, user mode
    g0.y = lds_addr;                                  // LDS byte address
    g0.z = (unsigned)(ga & 0xFFFFFFFFull);            // global_addr[31:0]
    g0.w = (unsigned)((ga >> 32) & 0x1FFFFFFull) | (2u << 30); // addr[56:32] | type=2
    const unsigned w0 = (2u << 16)                    // data_size = 4 bytes
                      | (1u << 20)                    // pad_enable
                      | (pad_interval_code << 22)
                      | (pad_amount_code << 25);
    const unsigned w1 = (tensor_d0 & 0xFFFFu) << 16;                   // dim0[15:0]
    const unsigned w2 = (tensor_d0 >> 16) | ((tensor_d1 & 0xFFFFu) << 16);
    const unsigned w3 = (tensor_d1 >> 16) | (tile_d0 << 16);
    const unsigned w4 = (tile_d1 & 0xFFFFu);                           // tile_dim2=0
    const unsigned w5 = (unsigned)(stride0 & 0xFFFFFFFFull);
    const unsigned w6 = (unsigned)((stride0 >> 32) & 0xFFFFull);       // dim1_stride=0
    v8i g1;
    g1[0] = (int)w0; g1[1] = (int)w1; g1[2] = (int)w2; g1[3] = (int)w3;
    g1[4] = (int)w4; g1[5] = (int)w5; g1[6] = (int)w6; g1[7] = 0;
    v4i gz4 = {0, 0, 0, 0};
    v8i gz8 = {0, 0, 0, 0, 0, 0, 0, 0};
    __builtin_amdgcn_tensor_load_to_lds(g0, g1, gz4, gz4, gz8, 0);
}

// ---------------- fp32 WMMA GEMM with TDM-staged, double-buffered LDS tiles --------
// Block: 256 threads = 8 waves; tile 128(M) x 64(N). Wave w: rows 16w..16w+15,
// 4 accumulators (16x16) spanning the 64 cols, A-fragment reused across them.
// TDM pads: A rows KC0 dwords + 2 (stride KC0+2, <=2-way LDS conflicts, 8B-aligned
// float2 frags); B rows 64 dwords + 8 (stride 72 -> half-waves on disjoint banks).
#define GEMM_TM 128
#define GEMM_TN 64
#define A_TILE_F (128 * 34)   // worst-case A buffer (floats)
#define B_TILE_F (32 * 72)    // B buffer (floats)

__global__ __launch_bounds__(256) void gemm_wmma_tdm_kernel(
    const float* __restrict__ A, const float* __restrict__ B,
    const float* __restrict__ bias, float* __restrict__ C,
    int M, int K, int N)
{
    __shared__ float smem[2 * A_TILE_F + 2 * B_TILE_F + 32];

    const int tid  = threadIdx.x;
    const int wave = tid >> 5;
    const int lane = tid & 31;
    const int lh   = lane & 15;
    const int ksel = (lane >> 4) << 1;       // 0 or 2

    const int rowBase0 = blockIdx.x * GEMM_TM;
    const int rowBase  = rowBase0 + wave * 16;
    const int colBase  = blockIdx.y * GEMM_TN;

    const int KC0     = (K < 32) ? K : 32;   // 8 or 32
    const int astride = KC0 + 2;
    const unsigned padiA = (KC0 == 8) ? 2u : 4u;  // pad after 8 / 32 dwords
    const int nch = K / KC0;

    // single runtime LDS base; all buffer addresses derived by scalar arithmetic
    float* smemf = &smem[0];
    const unsigned ldsBase = (unsigned)(size_t)smemf;
    const unsigned ldsA0 = ldsBase;
    const unsigned ldsA1 = ldsBase + A_TILE_F * 4u;
    const unsigned ldsB0 = ldsBase + 2u * A_TILE_F * 4u;
    const unsigned ldsB1 = ldsB0 + B_TILE_F * 4u;

    v8f acc[4];
#pragma unroll
    for (int j = 0; j < 4; ++j)
#pragma unroll
        for (int r = 0; r < 8; ++r) acc[j][r] = 0.0f;

    // prologue: wave 0 DMAs chunk 0 (TDM ignores EXEC; branch is wave-uniform)
    if (tid < 32) {
        tdm_load_2d(ldsA0, A + (size_t)rowBase0 * K,
                    (unsigned)K, (unsigned)(M - rowBase0),        // OOB rows read 0
                    (unsigned)KC0, GEMM_TM, (unsigned long long)K, padiA, 1u);
        tdm_load_2d(ldsB0, B + colBase,
                    (unsigned)(N - colBase), (unsigned)K,
                    GEMM_TN, (unsigned)KC0, (unsigned long long)N, 5u, 7u);
    }

    for (int ch = 0; ch < nch; ++ch) {
        const int cur = ch & 1;
        if (tid < 32) __builtin_amdgcn_s_wait_tensorcnt(0);
        __syncthreads();                      // tile `ch` visible to all waves
        if (tid < 32 && (ch + 1) < nch) {     // overlap next DMA with this chunk's math
            const int k1 = (ch + 1) * KC0;
            tdm_load_2d(cur ? ldsA0 : ldsA1, A + (size_t)rowBase0 * K + k1,
                        (unsigned)(K - k1), (unsigned)(M - rowBase0),
                        (unsigned)KC0, GEMM_TM, (unsigned long long)K, padiA, 1u);
            tdm_load_2d(cur ? ldsB0 : ldsB1, B + (size_t)k1 * N + colBase,
                        (unsigned)(N - colBase), (unsigned)(K - k1),
                        GEMM_TN, (unsigned)KC0, (unsigned long long)N, 5u, 7u);
        }

        const float* __restrict__ Acur = smemf + (cur ? A_TILE_F : 0);
        const float* __restrict__ Bt   = smemf + 2 * A_TILE_F + (cur ? B_TILE_F : 0);
        const float* __restrict__ Arow = Acur + (wave * 16 + lh) * astride;
        for (int ks = 0; ks < KC0; ks += 4) {
            const float2 av = *reinterpret_cast<const float2*>(Arow + ks + ksel);
            v2f a; a.x = av.x; a.y = av.y;
#pragma unroll
            for (int j = 0; j < 4; ++j) {
                const int c = j * 16 + lh;
                v2f b;
                b.x = Bt[(ks + ksel)     * 72 + c];
                b.y = Bt[(ks + ksel + 1) * 72 + c];
                acc[j] = __builtin_amdgcn_wmma_f32_16x16x4_f32(
                    false, a, false, b, (short)0, acc[j], false, false);
            }
        }
        __syncthreads();                      // done reading before buffer reuse
    }

    // C layout: VGPR r -> row = r + (lane>=16 ? 8 : 0), col = lane&15
    const int rofs = (lane >> 4) << 3;
    if (rowBase + 15 < M) {
#pragma unroll
        for (int j = 0; j < 4; ++j) {
            const int col = colBase + j * 16 + lh;
            const float bv = bias ? bias[col] : 0.0f;
#pragma unroll
            for (int r = 0; r < 8; ++r)
                C[(size_t)(rowBase + r + rofs) * N + col] = acc[j][r] + bv;
        }
    } else {
#pragma unroll
        for (int j = 0; j < 4; ++j) {
            const int col = colBase + j * 16 + lh;
            const float bv = bias ? bias[col] : 0.0f;
#pragma unroll
            for (int r = 0; r < 8; ++r) {
                const int row = rowBase + r + rofs;
                if (row < M) C[(size_t)row * N + col] = acc[j][r] + bv;
            }
        }
    }
}

// ---------------- attention coefficient dots: alpha_s/alpha_d [N,H] ----------------
__global__ void alpha_dots_kernel(const float* __restrict__ h,
                                  const float* __restrict__ a_src,
                                  const float* __restrict__ a_dst,
                                  float* __restrict__ as_, float* __restrict__ ad_,
                                  int N, int H, int C)
{
    long i = (long)blockIdx.x * blockDim.x + threadIdx.x;
    if (i >= (long)N * H) return;
    const int n = (int)(i / H), hh = (int)(i % H);
    const float* __restrict__ hr  = h + ((size_t)n * H + hh) * C;
    const float* __restrict__ asr = a_src + (size_t)hh * C;
    const float* __restrict__ adr = a_dst + (size_t)hh * C;
    float s = 0.f, d = 0.f;
    for (int c = 0; c < C; c += 4) {
        const float4 hv = *(const float4*)(hr + c);
        const float4 sv = *(const float4*)(asr + c);
        const float4 dv = *(const float4*)(adr + c);
        s += hv.x * sv.x + hv.y * sv.y + hv.z * sv.z + hv.w * sv.w;
        d += hv.x * dv.x + hv.y * dv.y + hv.z * dv.z + hv.w * dv.w;
    }
    as_[i] = s; ad_[i] = d;
}

// ---------------- edge pass 1: e = leaky_relu(as[src]+ad[dst]); segment max -------
__global__ void edge_e_max_kernel(const int* __restrict__ src, const int* __restrict__ dst,
                                  const float* __restrict__ as_, const float* __restrict__ ad_,
                                  float* __restrict__ ebuf, unsigned* __restrict__ mord,
                                  int E, int H)
{
    long i = (long)blockIdx.x * blockDim.x + threadIdx.x;
    if (i >= (long)E * H) return;
    const int e = (int)(i / H), hh = (int)(i % H);
    const int s = src[e], d = dst[e];
    float v = as_[s * H + hh] + ad_[d * H + hh];
    v = (v > 0.f) ? v : 0.2f * v;
    ebuf[i] = v;
    atomicMax(&mord[d * H + hh], f2ord(v));
}

// ---------------- edge pass 2: ex = exp(e - m[dst]); segment sum ------------------
__global__ void edge_exp_sum_kernel(const int* __restrict__ dst,
                                    float* __restrict__ ebuf,
                                    const unsigned* __restrict__ mord,
                                    float* __restrict__ denom,
                                    int E, int H)
{
    long i = (long)blockIdx.x * blockDim.x + threadIdx.x;
    if (i >= (long)E * H) return;
    const int e = (int)(i / H), hh = (int)(i % H);
    const int d = dst[e];
    const float ex = __expf(ebuf[i] - ord2f(mord[d * H + hh]));
    ebuf[i] = ex;
    atomicAdd(&denom[d * H + hh], ex);
}

// ---------------- edge pass 3: alpha = ex/denom; g[dst] += alpha*h[src] -----------
// One block per edge; threads cover (head, 4-channel group). H*(C/4) <= 256.
__global__ __launch_bounds__(256) void edge_scatter_kernel(
    const int* __restrict__ src, const int* __restrict__ dst,
    const float* __restrict__ h, const float* __restrict__ ebuf,
    const float* __restrict__ denom, float* __restrict__ g,
    float* __restrict__ att,
    int E, int H, int C, int gcols, int ghstride)
{
    const int e  = blockIdx.x;
    const int cg = C >> 2;
    const int t  = threadIdx.x;
    if (t >= H * cg) return;
    const int hh = t / cg, c4 = t % cg;
    const int s = src[e], d = dst[e];
    const float alpha = ebuf[(long)e * H + hh] / (denom[d * H + hh] + 1e-16f);
    if (att && c4 == 0) att[(long)e * H + hh] = alpha;
    const float4 hv = *(const float4*)(h + ((size_t)s * H + hh) * C + c4 * 4);
    float* gp = g + (size_t)d * gcols + hh * ghstride + c4 * 4;
    atomicAdd(gp + 0, hv.x * alpha);
    atomicAdd(gp + 1, hv.y * alpha);
    atomicAdd(gp + 2, hv.z * alpha);
    atomicAdd(gp + 3, hv.w * alpha);
}

// ---------------- x = relu(g*gscale + bgat + lin) ---------------------------------
__global__ void combine_relu_kernel(const float* __restrict__ g,
                                    const float* __restrict__ bgat,
                                    const float* __restrict__ lin,
                                    float* __restrict__ x, long n, int F, float gscale)
{
    long i = (long)blockIdx.x * blockDim.x + threadIdx.x;
    if (i >= n) return;
    const int f = (int)(i % F);
    const float v = g[i] * gscale + bgat[f] + lin[i];
    x[i] = v > 0.f ? v : 0.f;
}

// ---------------- BatchNorm stats (one block per channel) -------------------------
__global__ __launch_bounds__(256) void bn_stats_kernel(const float* __restrict__ x,
                                                       float* __restrict__ mu,
                                                       float* __restrict__ var,
                                                       int N, int F)
{
    __shared__ float sh[256], sh2[256];
    const int c = blockIdx.x, t = threadIdx.x;
    float s = 0.f, s2 = 0.f;
    for (int n = t; n < N; n += 256) {
        const float v = x[(size_t)n * F + c];
        s += v; s2 += v * v;
    }
    sh[t] = s; sh2[t] = s2;
    __syncthreads();
    for (int o = 128; o > 0; o >>= 1) {
        if (t < o) { sh[t] += sh[t + o]; sh2[t] += sh2[t + o]; }
        __syncthreads();
    }
    if (t == 0) {
        const float m = sh[0] / N;
        mu[c]  = m;
        var[c] = sh2[0] / N - m * m;
    }
}

// ---------------- BN apply + segment-max pool -------------------------------------
__global__ void bn_pool_kernel(const float* __restrict__ x,
                               const float* __restrict__ mu, const float* __restrict__ var,
                               const float* __restrict__ gamma, const float* __restrict__ beta,
                               const int* __restrict__ batch,
                               unsigned* __restrict__ pord, int N, int F)
{
    long i = (long)blockIdx.x * blockDim.x + threadIdx.x;
    if (i >= (long)N * F) return;
    const int n = (int)(i / F), c = (int)(i % F);
    const float y = (x[i] - mu[c]) * rsqrtf(var[c] + 1e-5f) * gamma[c] + beta[c];
    atomicMax(&pord[batch[n] * F + c], f2ord(y));
}

__global__ void pool_finalize_kernel(const unsigned* __restrict__ pord,
                                     float* __restrict__ out, int n)
{
    int i = blockIdx.x * blockDim.x + threadIdx.x;
    if (i < n) out[i] = ord2f(pord[i]);
}

// ===================================================================================
extern "C" void kernel_launch(void* const* d_in, const int* in_sizes, int n_in,
                              void* d_out, int out_size, void* d_ws, size_t ws_size,
                              hipStream_t stream)
{
    const int N = N_NODES, E = N_EDGES;
    const float* x0   = (const float*)d_in[0];
    const int*   eidx = (const int*)d_in[1];
    const int*   batch= (const int*)d_in[2];
    const float* W1 = (const float*)d_in[3],  *as1 = (const float*)d_in[4],
               * ad1 = (const float*)d_in[5], *bg1 = (const float*)d_in[6],
               * lw1 = (const float*)d_in[7], *lb1 = (const float*)d_in[8];
    const float* W2 = (const float*)d_in[9],  *as2 = (const float*)d_in[10],
               * ad2 = (const float*)d_in[11],*bg2 = (const float*)d_in[12],
               * lw2 = (const float*)d_in[13],*lb2 = (const float*)d_in[14];
    const float* W3 = (const float*)d_in[15], *as3 = (const float*)d_in[16],
               * ad3 = (const float*)d_in[17],*bg3 = (const float*)d_in[18],
               * lw3 = (const float*)d_in[19],*lb3 = (const float*)d_in[20];
    const float* bn_g = (const float*)d_in[21], *bn_b = (const float*)d_in[22];
    const int* src = eidx;
    const int* dst = eidx + E;

    // ---------------- workspace carve-up (256B aligned) ----------------
    char* ws = (char*)d_ws;
    size_t off = 0;
    auto carve = [&](size_t bytes) { void* p = ws + off; off += (bytes + 255) & ~(size_t)255; return p; };
    float*    hbuf  = (float*)   carve((size_t)N * 768 * 4); // h (max layer-3 width)
    float*    gbuf  = (float*)   carve((size_t)N * 256 * 4); // aggregated messages
    float*    xbuf  = (float*)   carve((size_t)N * 256 * 4); // current activations
    float*    linbf = (float*)   carve((size_t)N * 256 * 4); // linear residual
    float*    asb   = (float*)   carve((size_t)N * 6 * 4);
    float*    adb   = (float*)   carve((size_t)N * 6 * 4);
    unsigned* mord  = (unsigned*)carve((size_t)N * 6 * 4);
    float*    denom = (float*)   carve((size_t)N * 6 * 4);
    float*    ebuf  = (float*)   carve((size_t)E * 6 * 4);
    unsigned* pord  = (unsigned*)carve((size_t)N_GRAPHS * 128 * 4);
    float*    mu    = (float*)   carve(128 * 4);
    float*    var   = (float*)   carve(128 * 4);

    float* pooled_out = (float*)d_out;            // [64,128]
    float* att_out    = (float*)d_out + 64 * 128; // [E,6]

    auto g1d = [](long n) { return dim3((unsigned)((n + 255) / 256)); };

    auto gemm = [&](const float* A, const float* B, const float* bias, float* C,
                    int M, int K, int Nc) {
        dim3 grid((M + GEMM_TM - 1) / GEMM_TM, Nc / GEMM_TN);
        gemm_wmma_tdm_kernel<<<grid, 256, 0, stream>>>(A, B, bias, C, M, K, Nc);
    };

    auto gat_layer = [&](const float* xin, int Kin, const float* W, const float* a_s,
                         const float* a_d, const float* bgat, const float* lw,
                         const float* lb, int H, int C, bool concat, float* attw,
                         float* lin_out, int lin_cols, float* x_out) {
        const int HC = H * C;
        const int gcols = concat ? HC : C;
        gemm(xin, W, nullptr, hbuf, N, Kin, HC);
        gemm(xin, lw, lb, lin_out, N, Kin, lin_cols);
        alpha_dots_kernel<<<g1d((long)N * H), 256, 0, stream>>>(hbuf, a_s, a_d, asb, adb, N, H, C);
        fill_u32_kernel<<<g1d((long)N * H), 256, 0, stream>>>(mord, (long)N * H, ORD_NEG_INF);
        fill_f32_kernel<<<g1d((long)N * H), 256, 0, stream>>>(denom, (long)N * H, 0.f);
        fill_f32_kernel<<<g1d((long)N * gcols), 256, 0, stream>>>(gbuf, (long)N * gcols, 0.f);
        edge_e_max_kernel<<<g1d((long)E * H), 256, 0, stream>>>(src, dst, asb, adb, ebuf, mord, E, H);
        edge_exp_sum_kernel<<<g1d((long)E * H), 256, 0, stream>>>(dst, ebuf, mord, denom, E, H);
        edge_scatter_kernel<<<dim3(E), 256, 0, stream>>>(src, dst, hbuf, ebuf, denom, gbuf,
                                                         attw, E, H, C,
                                                         gcols, concat ? C : 0);
        combine_relu_kernel<<<g1d((long)N * gcols), 256, 0, stream>>>(
            gbuf, bgat, lin_out, x_out, (long)N * gcols, gcols, concat ? 1.0f : (1.0f / H));
    };

    // Layer 1: 8 -> 4x64 concat, residual lin1 (8->256)
    gat_layer(x0, 8, W1, as1, ad1, bg1, lw1, lb1, 4, 64, true, nullptr, linbf, 256, xbuf);
    // Layer 2: 256 -> 4x64 concat, residual lin2 (256->256)
    gat_layer(xbuf, 256, W2, as2, ad2, bg2, lw2, lb2, 4, 64, true, nullptr, linbf, 256, xbuf);
    // Layer 3: 256 -> 6x128 mean, residual lin3 (256->128); attention coeffs -> att_out
    gat_layer(xbuf, 256, W3, as3, ad3, bg3, lw3, lb3, 6, 128, false, att_out, linbf, 128, xbuf);

    // BatchNorm (training stats) + global max pool per graph
    bn_stats_kernel<<<dim3(128), 256, 0, stream>>>(xbuf, mu, var, N, 128);
    fill_u32_kernel<<<g1d(N_GRAPHS * 128), 256, 0, stream>>>(pord, N_GRAPHS * 128, ORD_NEG_INF);
    bn_pool_kernel<<<g1d((long)N * 128), 256, 0, stream>>>(xbuf, mu, var, bn_g, bn_b, batch, pord, N, 128);
    pool_finalize_kernel<<<g1d(N_GRAPHS * 128), 256, 0, stream>>>(pord, pooled_out, N_GRAPHS * 128);
}